// EctLayer_18545668784265
// MI455X (gfx1250) — compile-verified
//
#include <hip/hip_runtime.h>

// ECT layer for MI455X (gfx1250).
//  x:[100000,3] f32, v:[3,32] f32, lin:[32] f32, index:[100000] i32
//  out:[32,32,32] f32 (per-cloud max-normalized)
//
// Pipeline:
//  1) zero 128KB workspace accumulator
//  2) main kernel: heights via V_WMMA_F32_16X16X4_F32 (K=3 zero-padded to 4),
//     sigmoid filtration accumulated into a per-workgroup 128KB LDS histogram
//     (ds_add_f32), flushed once via global_atomic_add_f32
//  3) per-cloud max reduction + normalize

typedef __attribute__((ext_vector_type(2))) float v2f;
typedef __attribute__((ext_vector_type(8))) float v8f;

#define N_PTS   100000
#define BSEG    32
#define RRES    32
#define TTH     32
#define ECT_ELEMS (BSEG * RRES * TTH)      // 32768
#define NTILES  (N_PTS / 16)               // 6250 (exact)
#define SCALEF  100.0f
#define LOG2E   1.44269504088896340736f

#define WG      256
#define NWAVES  (WG / 32)
#define GRID    128

__global__ __launch_bounds__(WG) void ect_zero_ws(float* __restrict__ ws) {
    int i = blockIdx.x * WG + threadIdx.x;
    if (i < ECT_ELEMS) ws[i] = 0.0f;
}

__global__ __launch_bounds__(WG) void ect_accum(const float* __restrict__ x,
                                                const float* __restrict__ v,
                                                const float* __restrict__ lin,
                                                const int*   __restrict__ index,
                                                float*       __restrict__ ws) {
    // Full [B][R][T] f32 accumulator: 128 KB LDS (CDNA5 WGP has 320 KB).
    __shared__ float s_ect[ECT_ELEMS];
    for (int i = threadIdx.x; i < ECT_ELEMS; i += WG) s_ect[i] = 0.0f;
    __syncthreads();

    const int  lane  = threadIdx.x & 31;
    const int  l15   = lane & 15;
    const bool hi    = lane >= 16;
    const int  wave  = threadIdx.x >> 5;
    const int  gwave = blockIdx.x * NWAVES + wave;
    const int  nwav  = gridDim.x * NWAVES;

    // B fragments for theta groups 0 (t=0..15) and 1 (t=16..31).
    // B 4x16 f32 layout: lanes 0-15 hold K=0 (V0), K=1 (V1); lanes 16-31 hold
    // K=2 (V0), K=3 (V1). K=3 row is the zero pad.
    v2f bmat[2];
#pragma unroll
    for (int g = 0; g < 2; ++g) {
        const int t = g * 16 + l15;
        bmat[g].x = hi ? v[2 * TTH + t] : v[0 * TTH + t];
        bmat[g].y = hi ? 0.0f           : v[1 * TTH + t];
    }

    // sigmoid(SCALE*(lin[r]-nh)) = rcp(1 + exp2(LOG2E*SCALE*nh - LOG2E*SCALE*lin[r]))
    float larg[RRES];
#pragma unroll
    for (int r = 0; r < RRES; ++r) larg[r] = (-LOG2E * SCALEF) * lin[r];

    for (int tile = gwave; tile < NTILES; tile += nwav) {
        const int pbase = tile * 16;

        // A fragment (16x4 f32): lanes 0-15: (x0,x1) of point pbase+l15;
        // lanes 16-31: (x2, 0) of the same point.
        v2f a;
        {
            const float* xp = x + 3 * (pbase + l15);
            a.x = hi ? xp[2] : xp[0];
            a.y = hi ? 0.0f  : xp[1];
        }

        // Segment ids for this lane's D rows: M = k (+8 for hi half), k=0..7.
        int myidx[8];
        {
            const int* ip = index + pbase + (hi ? 8 : 0);   // 32B aligned
            const int4 i0 = *(const int4*)(ip);
            const int4 i1 = *(const int4*)(ip + 4);
            myidx[0] = i0.x; myidx[1] = i0.y; myidx[2] = i0.z; myidx[3] = i0.w;
            myidx[4] = i1.x; myidx[5] = i1.y; myidx[6] = i1.z; myidx[7] = i1.w;
        }

#pragma unroll
        for (int g = 0; g < 2; ++g) {
            v8f c = {};
            // nh for 16 points x 16 thetas: D = A(16x4) * B(4x16) + 0
            v8f d = __builtin_amdgcn_wmma_f32_16x16x4_f32(
                false, a, false, bmat[g], (short)0, c, false, false);
            const int tcol = g * 16 + l15;
#pragma unroll
            for (int k = 0; k < 8; ++k) {
                const float h = (LOG2E * SCALEF) * d[k];
                float* base = &s_ect[myidx[k] * (RRES * TTH) + tcol];
#pragma unroll
                for (int r = 0; r < RRES; ++r) {
                    const float e = __builtin_amdgcn_exp2f(h + larg[r]);
                    const float s = __builtin_amdgcn_rcpf(1.0f + e);
                    unsafeAtomicAdd(base + r * TTH, s);   // ds_add_f32
                }
            }
        }
    }

    __syncthreads();
    for (int i = threadIdx.x; i < ECT_ELEMS; i += WG)
        unsafeAtomicAdd(&ws[i], s_ect[i]);                // global_atomic_add_f32
}

__global__ __launch_bounds__(WG) void ect_normalize(const float* __restrict__ ws,
                                                    float*       __restrict__ out) {
    __shared__ float red[WG];
    const int    b = blockIdx.x;
    const float* p = ws + b * (RRES * TTH);

    float m = -INFINITY;
    for (int i = threadIdx.x; i < RRES * TTH; i += WG) m = fmaxf(m, p[i]);
    red[threadIdx.x] = m;
    __syncthreads();
#pragma unroll
    for (int s = WG / 2; s > 0; s >>= 1) {
        if (threadIdx.x < s)
            red[threadIdx.x] = fmaxf(red[threadIdx.x], red[threadIdx.x + s]);
        __syncthreads();
    }
    const float mx = red[0];
    for (int i = threadIdx.x; i < RRES * TTH; i += WG)
        out[b * (RRES * TTH) + i] = p[i] / mx;
}

extern "C" void kernel_launch(void* const* d_in, const int* in_sizes, int n_in,
                              void* d_out, int out_size, void* d_ws, size_t ws_size,
                              hipStream_t stream) {
    const float* x     = (const float*)d_in[0];
    const float* v     = (const float*)d_in[1];
    const float* lin   = (const float*)d_in[2];
    const int*   index = (const int*)  d_in[3];
    float*       out   = (float*)d_out;
    float*       ws    = (float*)d_ws;   // needs ECT_ELEMS floats (128 KB)

    ect_zero_ws<<<(ECT_ELEMS + WG - 1) / WG, WG, 0, stream>>>(ws);
    ect_accum<<<GRID, WG, 0, stream>>>(x, v, lin, index, ws);
    ect_normalize<<<BSEG, WG, 0, stream>>>(ws, out);
}